// loss_39797166965275
// MI455X (gfx1250) — compile-verified
//
#include <hip/hip_runtime.h>
#include <math.h>

// Problem constants (match reference)
#define B_N 8192
#define T_N 1024
#define K_N 4
#define C_N 16            // chunks per sequence for parallel scan
#define L_N (T_N / C_N)   // 64 timesteps per chunk
#define ACH 128           // numerator-kernel timesteps per thread

typedef float v2f __attribute__((ext_vector_type(2)));
typedef float v8f __attribute__((ext_vector_type(8)));

// ws layout: byte 0: double accumulator (total llh); byte 256+: chunk matrices,
// per (b, c): 16 floats P[k][j] (transfer j->k) + 4 floats per-column (basis j)
// log-scale -> 20 floats.

__global__ void crf_zero(double* acc) {
    if (threadIdx.x == 0 && blockIdx.x == 0) *acc = 0.0;
}

// ---------------- Phase A: gold-path (numerator) score, chunked over T ------
__global__ void crf_numer(const float* __restrict__ em, const int* __restrict__ tags,
                          const float* __restrict__ startT, const float* __restrict__ endT,
                          const float* __restrict__ trans, double* acc) {
    const int chunks = T_N / ACH;
    int tid = blockIdx.x * blockDim.x + threadIdx.x;
    int b = tid / chunks, c = tid % chunks;
    float sum = 0.f;
    if (b < B_N) {
        int s = c * ACH, e = s + ACH;
        const int* tg = tags + (size_t)b * T_N;
        const float* eb = em + (size_t)b * T_N * K_N;
        int tag = tg[s];
        if (s == 0) sum += startT[tag];
        for (int t = s; t < e; ++t) {
            sum += eb[t * K_N + tag];
            if (t < T_N - 1) {            // transition (t -> t+1), bridges chunks
                int nt = tg[t + 1];
                sum += trans[tag * K_N + nt];
                tag = nt;
            }
        }
        if (e == T_N) sum += endT[tag];
    }
    for (int off = 16; off; off >>= 1) sum += __shfl_down(sum, off, 32);
    if ((threadIdx.x & 31) == 0) atomicAdd(acc, (double)sum);
}

// ---------------- Phase B: 4x4 chunk transfer matrices via WMMA -------------
// One wave handles 4 batches x 1 chunk. State S is 4(prev state) x 16 cols
// (4 batches x 4 basis vectors). Step: D = A x S, A rows {0,1,8,9} hold
// expT^T rows for k'={0,1,2,3}; with D's VGPR layout (M=r lanes0-15,
// M=r+8 lanes16-31) d[0],d[1] land directly in B's 4x16 layout
// (K=0/2 in VGPR0, K=1/3 in VGPR1) -- no cross-lane shuffle needed.
// Columns then scaled by exp(emission). Scaled-prob domain with per-column
// log2-scale, renormalized every 8 steps.
__global__ void crf_chunkmat(const float* __restrict__ em,
                             const float* __restrict__ trans,
                             float* __restrict__ mats) {
    int lane = threadIdx.x & 31;
    int wid  = __builtin_amdgcn_readfirstlane(
                   blockIdx.x * (blockDim.x >> 5) + (threadIdx.x >> 5));
    int bg = wid / C_N;            // group of 4 batches (scalar)
    int ck = wid % C_N;            // chunk index (scalar)
    int row  = lane & 15;
    int half = lane >> 4;          // 0: K rows {0,1}; 1: K rows {2,3}
    int k0 = half ? 2 : 0;
    int k1 = k0 + 1;

    // A (16x4): rows 0,1 = expT^T rows k'=0,1 ; rows 8,9 = expT^T rows k'=2,3
    bool vld = (row < 2) | (row == 8) | (row == 9);
    int  kp  = vld ? ((row < 2) ? row : (row - 6)) : 0;
    v2f a;
    a.x = vld ? __expf(trans[k0 * 4 + kp]) : 0.f;
    a.y = vld ? __expf(trans[k1 * 4 + kp]) : 0.f;

    // B (4x16): column n=row -> (local batch bl, basis j); identity init
    int bl = row >> 2, j = row & 3;
    int b  = bg * 4 + bl;
    float b0 = (k0 == j) ? 1.f : 0.f;
    float b1 = (k1 == j) ? 1.f : 0.f;
    float lacc = 0.f;              // log2 of column scale

    // per-lane emission base: this lane always needs the (k0,k0+1) float pair
    const float* pbase = em + (size_t)b * T_N * K_N + k0;
    int t0 = ck * L_N + 1;                                   // scalar
    int t1 = (ck + 1) * L_N; if (t1 > T_N - 1) t1 = T_N - 1; // scalar

    auto step = [&](int t) {
        v2f e = *(const v2f*)(pbase + (size_t)t * K_N);  // one global_load_b64
        float e0 = __expf(e.x);
        float e1 = __expf(e.y);
        v2f bm; bm.x = b0; bm.y = b1;
        v8f cz = {};
        v8f d = __builtin_amdgcn_wmma_f32_16x16x4_f32(
            false, a, false, bm, (short)0, cz, false, false);
        b0 = d[0] * e0;            // d[0],d[1] are already in B layout
        b1 = d[1] * e1;
    };
    auto renorm = [&]() {
        float m = fmaxf(b0, b1);
        m = fmaxf(m, __shfl_xor(m, 16, 32));     // max over the 4 K values
        float inv = __builtin_amdgcn_rcpf(m);
        b0 *= inv; b1 *= inv;
        lacc += __log2f(m);
    };

    // 7 unconditional blocks of 8 steps (t0..t0+55 always < T)
    for (int blk = 0; blk < 7; ++blk) {
        int tb = t0 + blk * 8;
#pragma unroll
        for (int u = 0; u < 8; ++u) step(tb + u);
        renorm();
    }
    // final block: up to 8 steps, scalar-guarded (last chunk has 63 steps)
    {
        int tb = t0 + 56;
#pragma unroll
        for (int u = 0; u < 8; ++u) { int t = tb + u; if (t <= t1) step(t); }
        renorm();
    }
    lacc *= 0.6931471805599453f;   // log2 -> ln

    float* out = mats + ((size_t)b * C_N + ck) * 20;
    out[k0 * 4 + j] = b0;          // P[k0][j]
    out[k1 * 4 + j] = b1;          // P[k1][j]
    out[16 + j]     = lacc;        // ls[j]; lanes l and l+16 write same value
}

// ---------------- Phase C: per-batch combine + denominator ------------------
__global__ void crf_denom(const float* __restrict__ em,
                          const float* __restrict__ startT,
                          const float* __restrict__ endT,
                          const float* __restrict__ mats, double* acc) {
    int b = blockIdx.x * blockDim.x + threadIdx.x;
    float contrib = 0.f;
    if (b < B_N) {
        float tv[4], v[4];
        float mx = -1e30f;
        for (int k = 0; k < 4; ++k) {
            tv[k] = startT[k] + em[(size_t)b * T_N * K_N + k];
            mx = fmaxf(mx, tv[k]);
        }
        for (int k = 0; k < 4; ++k) v[k] = __expf(tv[k] - mx);
        float lognorm = mx;
        const float* mb = mats + (size_t)b * C_N * 20;
        for (int c = 0; c < C_N; ++c) {
            const float* M = mb + c * 20;
            const float* ls = M + 16;
            float lsmax = fmaxf(fmaxf(ls[0], ls[1]), fmaxf(ls[2], ls[3]));
            float u[4];
            for (int jj = 0; jj < 4; ++jj)
                u[jj] = __expf(ls[jj] - lsmax) * v[jj];
            float s[4]; float m2 = 0.f;
            for (int k = 0; k < 4; ++k) {   // alpha_out[k] = sum_j P[k][j] u[j]
                s[k] = M[k * 4 + 0] * u[0] + M[k * 4 + 1] * u[1]
                     + M[k * 4 + 2] * u[2] + M[k * 4 + 3] * u[3];
                m2 = fmaxf(m2, s[k]);
            }
            float inv = __builtin_amdgcn_rcpf(m2);
            for (int k = 0; k < 4; ++k) v[k] = s[k] * inv;
            lognorm += lsmax + __logf(m2);
        }
        float se = 0.f;
        for (int k = 0; k < 4; ++k) se += v[k] * __expf(endT[k]);
        contrib = -(lognorm + __logf(se));   // -denominator_b
    }
    for (int off = 16; off; off >>= 1) contrib += __shfl_down(contrib, off, 32);
    if ((threadIdx.x & 31) == 0) atomicAdd(acc, (double)contrib);
}

// ---------------- Phase D: focal transform of the scalar NLL ----------------
__global__ void crf_final(const double* __restrict__ acc, float* __restrict__ out) {
    if (threadIdx.x == 0 && blockIdx.x == 0) {
        double llh  = *acc;          // sum_b (numer_b - denom_b)
        double logp = -llh;
        double p    = exp(llh);
        double om   = 1.0 - p;
        out[0] = (float)(om * om * logp);   // GAMMA = 2
    }
}

extern "C" void kernel_launch(void* const* d_in, const int* in_sizes, int n_in,
                              void* d_out, int out_size, void* d_ws, size_t ws_size,
                              hipStream_t stream) {
    const float* em     = (const float*)d_in[0];   // [B,T,K]
    const int*   tags   = (const int*)d_in[1];     // [B,T]
    const float* startT = (const float*)d_in[2];   // [K]
    const float* endT   = (const float*)d_in[3];   // [K]
    const float* trans  = (const float*)d_in[4];   // [K,K]
    float* out = (float*)d_out;

    double* acc  = (double*)d_ws;
    float*  mats = (float*)((char*)d_ws + 256);    // B*C*20 floats (~10.5 MB)

    crf_zero<<<1, 32, 0, stream>>>(acc);

    {   // Phase A: numerator (also warms L2 with emissions/tags)
        int threads = B_N * (T_N / ACH);           // 65536
        crf_numer<<<threads / 256, 256, 0, stream>>>(em, tags, startT, endT, trans, acc);
    }
    {   // Phase B: WMMA chunk matrices; (B/4)*C waves, 8 waves per block
        int waves  = (B_N / 4) * C_N;              // 32768
        int blocks = waves / 8;                    // 4096
        crf_chunkmat<<<blocks, 256, 0, stream>>>(em, trans, mats);
    }
    {   // Phase C: per-batch combine + denominator
        crf_denom<<<B_N / 256, 256, 0, stream>>>(em, startT, endT, mats, acc);
    }
    crf_final<<<1, 32, 0, stream>>>(acc, out);
}